// DepConv3D_60198261620899
// MI455X (gfx1250) — compile-verified
//
#include <hip/hip_runtime.h>

typedef __attribute__((ext_vector_type(2))) float v2f;
typedef __attribute__((ext_vector_type(8))) float v8f;

#define DIM_H 512
#define DIM_W 512
#define IC 16
#define OC 32
#define NB 4

// Block: 256 threads = 8 waves. Each wave: one output row, 16 pixels wide,
// all 32 output channels (two 16x16x4-f32 WMMA accumulator tiles).
__global__ __launch_bounds__(256)
void depconv3d_wmma_kernel(const float* __restrict__ feat,
                           const int*   __restrict__ depth,
                           const float* __restrict__ wt,
                           float*       __restrict__ out) {
  // Weights pre-swizzled for WMMA A layout:
  //   wldsf[((((tap*2+d)*4+c)*2+half)*32 + o)*2 + j] = wt[o, i=4c+2*half+j, d, tap]
  __shared__ float wldsf[9 * 2 * 4 * 2 * 32 * 2];        // 36864 B
  // Feature halo tile, channel-fastest: fldsf[r][col][i], r in [0,10), col in [0,18)
  __shared__ float fldsf[10 * 18 * IC];                  // 11520 B
  __shared__ int   dlds[10 * 18];                        //   720 B

  const int tid = threadIdx.x;
  const int b  = blockIdx.z;
  const int h0 = blockIdx.y * 8;
  const int w0 = blockIdx.x * 16;

  // ---- cooperative weight stage (d=2 slice of the 3x3x3 kernel is never selected) ----
  for (int e = tid; e < 9216; e += 256) {
    const int j    = e & 1;
    const int o    = (e >> 1) & 31;
    const int half = (e >> 6) & 1;
    const int c    = (e >> 7) & 3;
    const int d    = (e >> 9) & 1;
    const int tap  = e >> 10;
    const int i    = 4 * c + 2 * half + j;
    // weight layout (oC, iC, 3, 3, 3): o*432 + i*27 + d*9 + tap
    wldsf[e] = wt[o * 432 + i * 27 + d * 9 + tap];
  }
  // ---- cooperative feature halo stage (zero padding, matches F.unfold) ----
  for (int e = tid; e < 10 * 18 * IC; e += 256) {
    const int i  = e & (IC - 1);
    const int cc = (e >> 4) % 18;
    const int r  = e / (18 * IC);
    const int hh = h0 + r - 1;
    const int ww = w0 + cc - 1;
    float v = 0.f;
    if (hh >= 0 && hh < DIM_H && ww >= 0 && ww < DIM_W)
      v = feat[((b * IC + i) * DIM_H + hh) * DIM_W + ww];
    fldsf[e] = v;
  }
  // ---- cooperative depth halo stage (padding depth = 0, matches reference) ----
  for (int e = tid; e < 180; e += 256) {
    const int cc = e % 18;
    const int r  = e / 18;
    const int hh = h0 + r - 1;
    const int ww = w0 + cc - 1;
    int v = 0;
    if (hh >= 0 && hh < DIM_H && ww >= 0 && ww < DIM_W)
      v = depth[(b * DIM_H + hh) * DIM_W + ww];
    dlds[e] = v;
  }
  __syncthreads();

  const int lane = tid & 31;
  const int wv   = tid >> 5;     // wave id = row within tile
  const int m    = lane & 15;    // pixel column / output-channel row (WMMA lane group)
  const int half = lane >> 4;

  const v2f* wlds2 = (const v2f*)wldsf;
  const int dc = dlds[(wv + 1) * 18 + (m + 1)];   // center depth for this pixel

  v8f acc0 = {};  // output channels 0..15
  v8f acc1 = {};  // output channels 16..31

  for (int tap = 0; tap < 9; ++tap) {
    const int dh = tap / 3;
    const int dw = tap - 3 * dh;
    const int dn   = dlds[(wv + dh) * 18 + (m + dw)];
    const int diff = dn - dc;
    const int fbase = ((wv + dh) * 18 + (m + dw)) * IC + 2 * half;
    v2f fv[4];
#pragma unroll
    for (int c = 0; c < 4; ++c)
      fv[c] = *(const v2f*)&fldsf[fbase + 4 * c];   // ds_load_b64, K-pair for this lane
#pragma unroll
    for (int d = 0; d < 2; ++d) {
      // gate: d=0 selected when diff==-1, d=1 when diff==0, else contributes zero
      const bool g = (diff == d - 1);
#pragma unroll
      for (int c = 0; c < 4; ++c) {
        v2f bv;
        bv.x = g ? fv[c].x : 0.f;
        bv.y = g ? fv[c].y : 0.f;
        const int wbase = (((tap * 2 + d) * 4 + c) * 2 + half) * 32;
        const v2f a0 = wlds2[wbase + m];
        const v2f a1 = wlds2[wbase + 16 + m];
        acc0 = __builtin_amdgcn_wmma_f32_16x16x4_f32(
            false, a0, false, bv, (short)0, acc0, false, false);
        acc1 = __builtin_amdgcn_wmma_f32_16x16x4_f32(
            false, a1, false, bv, (short)0, acc1, false, false);
      }
    }
  }

  // D layout (16x16 f32): VGPR v holds M = v + 8*half, N = lane&15
  const int hh = h0 + wv;
  const int ww = w0 + m;
#pragma unroll
  for (int v = 0; v < 8; ++v) {
    const int o = v + 8 * half;
    out[((b * OC + o) * DIM_H + hh) * DIM_W + ww]        = acc0[v];
    out[((b * OC + 16 + o) * DIM_H + hh) * DIM_W + ww]   = acc1[v];
  }
}

extern "C" void kernel_launch(void* const* d_in, const int* in_sizes, int n_in,
                              void* d_out, int out_size, void* d_ws, size_t ws_size,
                              hipStream_t stream) {
  (void)in_sizes; (void)n_in; (void)out_size; (void)d_ws; (void)ws_size;
  const float* feat  = (const float*)d_in[0];
  const int*   depth = (const int*)d_in[1];
  const float* wt    = (const float*)d_in[2];
  float*       out   = (float*)d_out;
  dim3 grid(DIM_W / 16, DIM_H / 8, NB);
  depconv3d_wmma_kernel<<<grid, 256, 0, stream>>>(feat, depth, wt, out);
}